// GCNBlock_45947560133452
// MI455X (gfx1250) — compile-verified
//
#include <hip/hip_runtime.h>

#define D 128
#define LN_EPS 1e-5f

typedef __attribute__((ext_vector_type(2))) float v2f;
typedef __attribute__((ext_vector_type(8))) float v8f;

// ---------------- degree / dinv ----------------
__global__ void init_deg_kernel(float* __restrict__ deg, int N) {
  int i = blockIdx.x * blockDim.x + threadIdx.x;
  if (i < N) deg[i] = 1.0f;  // self-loop weight
}

__global__ void edge_deg_kernel(const int* __restrict__ ei, const float* __restrict__ w,
                                float* __restrict__ deg, int E) {
  int e = blockIdx.x * blockDim.x + threadIdx.x;
  if (e < E) atomicAdd(&deg[ei[E + e]], w[e]);  // col = ei[1][e]
}

__global__ void deg_to_dinv_kernel(float* __restrict__ deg, int N) {
  int i = blockIdx.x * blockDim.x + threadIdx.x;
  if (i < N) {
    float d = deg[i];
    deg[i] = d > 0.0f ? rsqrtf(d) : 0.0f;
  }
}

// ---------------- GEMM1: H = X @ W  (N x 128 @ 128 x 128) ----------------
// Block = 256 threads = 8 waves. Block covers 16 rows x 128 cols; wave w does
// the 16x16 tile at cols [16w, 16w+16). fp32 WMMA, K stepped by 4.
__global__ void __launch_bounds__(256)
gemm_xW_kernel(const float* __restrict__ X, const float* __restrict__ W,
               float* __restrict__ H) {
  const int wave = threadIdx.x >> 5;
  const int lane = threadIdx.x & 31;
  const int row0 = blockIdx.x * 16;
  const int col0 = wave * 16;
  const int m    = lane & 15;
  const int half = lane >> 4;          // 0 or 1 -> K pair 0..1 or 2..3
  const int n    = col0 + m;           // B/C column for this lane

  v8f acc = {};
  const float* xrow = X + (size_t)(row0 + m) * D + half * 2;
  for (int k = 0; k < D; k += 4) {
    v2f a = *(const v2f*)(xrow + k);                 // A[m][k+2h], A[m][k+2h+1]
    const float* wp = W + (size_t)(k + half * 2) * D + n;
    v2f b;
    b.x = wp[0];                                     // B[k+2h][n]
    b.y = wp[D];                                     // B[k+2h+1][n]
    acc = __builtin_amdgcn_wmma_f32_16x16x4_f32(
        false, a, false, b, (short)0, acc, false, false);
  }
  float* hp = H + (size_t)(row0 + half * 8) * D + n; // rows r + 8*half
#pragma unroll
  for (int r = 0; r < 8; ++r) hp[(size_t)r * D] = acc[r];
}

// ---------------- self-loop: agg = dinv^2 * h (also initializes agg) -------
__global__ void self_loop_kernel(const float* __restrict__ h, const float* __restrict__ dinv,
                                 float* __restrict__ agg, int N) {
  int t = blockIdx.x * blockDim.x + threadIdx.x;   // one thread per 4 features
  int node = t >> 5;
  int f = (t & 31) * 4;
  if (node < N) {
    float c = dinv[node];
    c = c * c;
    float4 hv = *(const float4*)(h + (size_t)node * D + f);
    float4 o;
    o.x = c * hv.x; o.y = c * hv.y; o.z = c * hv.z; o.w = c * hv.w;
    *(float4*)(agg + (size_t)node * D + f) = o;
  }
}

// ---------------- edge scatter: wave per edge, 4 feats per lane ------------
__global__ void __launch_bounds__(256)
edge_scatter_kernel(const int* __restrict__ ei, const float* __restrict__ w,
                    const float* __restrict__ dinv, const float* __restrict__ h,
                    float* __restrict__ agg, int E) {
  int wave = threadIdx.x >> 5;
  int lane = threadIdx.x & 31;
  int e = blockIdx.x * 8 + wave;
  if (e >= E) return;
  int r = ei[e];          // source
  int c = ei[E + e];      // target
  float coeff = dinv[r] * w[e] * dinv[c];
  int f = lane * 4;
  float4 hv = *(const float4*)(h + (size_t)r * D + f);
  float* ap = agg + (size_t)c * D + f;
  atomicAdd(ap + 0, coeff * hv.x);
  atomicAdd(ap + 1, coeff * hv.y);
  atomicAdd(ap + 2, coeff * hv.z);
  atomicAdd(ap + 3, coeff * hv.w);
}

// ---------------- bias + relu + LayerNorm: wave per node -------------------
__global__ void __launch_bounds__(256)
bias_relu_ln_kernel(const float* __restrict__ agg, const float* __restrict__ bias,
                    const float* __restrict__ gamma, const float* __restrict__ beta,
                    float* __restrict__ y, int N) {
  int wave = threadIdx.x >> 5;
  int lane = threadIdx.x & 31;
  int node = blockIdx.x * 8 + wave;
  if (node >= N) return;
  int f = lane * 4;
  float4 v = *(const float4*)(agg + (size_t)node * D + f);
  float4 b = *(const float4*)(bias + f);
  v.x = fmaxf(v.x + b.x, 0.0f);
  v.y = fmaxf(v.y + b.y, 0.0f);
  v.z = fmaxf(v.z + b.z, 0.0f);
  v.w = fmaxf(v.w + b.w, 0.0f);
  float s = v.x + v.y + v.z + v.w;
#pragma unroll
  for (int mski = 16; mski >= 1; mski >>= 1) s += __shfl_xor(s, mski, 32);
  float mu = s * (1.0f / D);
  float dx0 = v.x - mu, dx1 = v.y - mu, dx2 = v.z - mu, dx3 = v.w - mu;
  float vs = dx0 * dx0 + dx1 * dx1 + dx2 * dx2 + dx3 * dx3;
#pragma unroll
  for (int mski = 16; mski >= 1; mski >>= 1) vs += __shfl_xor(vs, mski, 32);
  float inv = rsqrtf(vs * (1.0f / D) + LN_EPS);
  float4 g  = *(const float4*)(gamma + f);
  float4 be = *(const float4*)(beta + f);
  float4 o;
  o.x = dx0 * inv * g.x + be.x;
  o.y = dx1 * inv * g.y + be.y;
  o.z = dx2 * inv * g.z + be.z;
  o.w = dx3 * inv * g.w + be.w;
  *(float4*)(y + (size_t)node * D + f) = o;
}

// ---------------- GEMM2: OUT = Y @ linW^T + lin_b --------------------------
// B[k][j] = linW[j*D + k]  => contiguous float2 loads per lane.
__global__ void __launch_bounds__(256)
gemm_y_linT_kernel(const float* __restrict__ Y, const float* __restrict__ linW,
                   const float* __restrict__ linb, float* __restrict__ OUT) {
  const int wave = threadIdx.x >> 5;
  const int lane = threadIdx.x & 31;
  const int row0 = blockIdx.x * 16;
  const int col0 = wave * 16;
  const int m    = lane & 15;
  const int half = lane >> 4;
  const int n    = col0 + m;

  float lb = linb[n];
  v8f acc = {lb, lb, lb, lb, lb, lb, lb, lb};
  const float* yrow = Y + (size_t)(row0 + m) * D + half * 2;
  const float* brow = linW + (size_t)n * D + half * 2;
  for (int k = 0; k < D; k += 4) {
    v2f a = *(const v2f*)(yrow + k);
    v2f b = *(const v2f*)(brow + k);   // linW[n][k+2h], linW[n][k+2h+1]
    acc = __builtin_amdgcn_wmma_f32_16x16x4_f32(
        false, a, false, b, (short)0, acc, false, false);
  }
  float* op = OUT + (size_t)(row0 + half * 8) * D + n;
#pragma unroll
  for (int r = 0; r < 8; ++r) op[(size_t)r * D] = acc[r];
}

// ---------------------------------------------------------------------------
extern "C" void kernel_launch(void* const* d_in, const int* in_sizes, int n_in,
                              void* d_out, int out_size, void* d_ws, size_t ws_size,
                              hipStream_t stream) {
  const float* x     = (const float*)d_in[0];
  const int*   ei    = (const int*)d_in[1];
  const float* ew    = (const float*)d_in[2];
  const float* W     = (const float*)d_in[3];
  const float* bias  = (const float*)d_in[4];
  const float* gamma = (const float*)d_in[5];
  const float* beta  = (const float*)d_in[6];
  const float* linW  = (const float*)d_in[7];
  const float* linb  = (const float*)d_in[8];
  float* out = (float*)d_out;

  const int N = in_sizes[0] / D;   // 100000 (multiple of 16)
  const int E = in_sizes[2];       // 1600000

  char* ws = (char*)d_ws;
  float* dinv = (float*)ws;
  size_t off = (((size_t)N * sizeof(float)) + 255) & ~(size_t)255;
  float* h   = (float*)(ws + off);
  float* agg = (float*)(ws + off + (size_t)N * D * sizeof(float));

  init_deg_kernel<<<(N + 255) / 256, 256, 0, stream>>>(dinv, N);
  edge_deg_kernel<<<(E + 255) / 256, 256, 0, stream>>>(ei, ew, dinv, E);
  deg_to_dinv_kernel<<<(N + 255) / 256, 256, 0, stream>>>(dinv, N);
  gemm_xW_kernel<<<N / 16, 256, 0, stream>>>(x, W, h);
  self_loop_kernel<<<((size_t)N * 32 + 255) / 256, 256, 0, stream>>>(h, dinv, agg, N);
  edge_scatter_kernel<<<(E + 7) / 8, 256, 0, stream>>>(ei, ew, dinv, h, agg, E);
  bias_relu_ln_kernel<<<(N + 7) / 8, 256, 0, stream>>>(agg, bias, gamma, beta, h, N);
  gemm_y_linT_kernel<<<N / 16, 256, 0, stream>>>(h, linW, linb, out);
}